// CRF_78151224918717
// MI455X (gfx1250) — compile-verified
//
#include <hip/hip_runtime.h>
#include <stdint.h>

#define LBL     50
#define LPAD    64
#define TMAX    1024
#define BATCH   512
#define START_S 49
#define STOP_S  48
#define NEGV    (-10000.0f)
#define TILE    128
#define NSEG    (TMAX / TILE)

// ---- CDNA5 async-to-LDS copy (ASYNCcnt path, CDNA5 ISA) ----
__device__ __forceinline__ void async_ld_b128(uint32_t lds_off, int voff,
                                              unsigned long long saddr) {
  // GVS mode: addr = SADDR(u64) + VADDR(i32); VDST holds LDS byte address.
  asm volatile("global_load_async_to_lds_b128 %0, %1, %2"
               :: "v"(lds_off), "v"(voff), "s"(saddr)
               : "memory");
}
__device__ __forceinline__ void wait_async0() {
  asm volatile("s_wait_asynccnt 0" ::: "memory");
}

// argmax over from=0..49 of vit[from] + tr[from]; pads (50,51) can never win
// because vit[50..63] == -1e30.  Strict '>' scanned ascending == jnp first-max.
__device__ __forceinline__ int argmax_row(const float* __restrict__ vit,
                                          const float (&tr)[52],
                                          float* __restrict__ best_out) {
  const float4* v4 = (const float4*)vit;
  float best = -3.0e38f;
  int bi = 0;
  #pragma unroll
  for (int c = 0; c < 13; ++c) {
    float4 vv = v4[c];
    float s0 = vv.x + tr[4 * c + 0]; if (s0 > best) { best = s0; bi = 4 * c + 0; }
    float s1 = vv.y + tr[4 * c + 1]; if (s1 > best) { best = s1; bi = 4 * c + 1; }
    float s2 = vv.z + tr[4 * c + 2]; if (s2 > best) { best = s2; bi = 4 * c + 2; }
    float s3 = vv.w + tr[4 * c + 3]; if (s3 > best) { best = s3; bi = 4 * c + 3; }
  }
  *best_out = best;
  return bi;
}

extern "C" __global__ __launch_bounds__(64)
void crf_viterbi_kernel(const float* __restrict__ feats,   // [B,T,L]
                        const int*   __restrict__ lens,    // [B]
                        const float* __restrict__ trans,   // [L,L]
                        float*       __restrict__ out,     // [B] scores ++ [B*T] paths
                        uint8_t*     __restrict__ ptrws)   // [B][T][LPAD]
{
  __shared__ __align__(16) float   vit_sh[2][LPAD];        // ping-pong => 1 barrier/step
  __shared__ __align__(16) uint8_t tile_sh[2][TILE * LPAD];
  __shared__ __align__(16) uint8_t path_sh[TMAX];
  __shared__ int idx_sh;

  const int  b      = blockIdx.x;
  const int  to     = threadIdx.x;         // one 'to' state per thread (50 active)
  const bool act    = (to < LBL);
  const int  tosafe = act ? to : (LBL - 1);  // clamp: inactive lanes mirror lane 49
  const int  len    = lens[b];             // 1..T, wave-uniform per block

  // Transition row for this 'to' state, kept in VGPRs for the whole scan.
  float tr[52];
  #pragma unroll
  for (int f = 0; f < LBL; ++f) tr[f] = trans[tosafe * LBL + f];
  tr[50] = 0.0f; tr[51] = 0.0f;
  const float tstop = trans[STOP_S * LBL + tosafe];

  // vit init: NEG everywhere, 0 at START; pads [50..63] = -1e30 in BOTH buffers,
  // never rewritten (act-guarded writes), so argmax can never select them.
  vit_sh[0][to] = act ? (to == START_S ? 0.0f : NEGV) : -1.0e30f;
  vit_sh[1][to] = -1.0e30f;                // real entries overwritten at t=0 before use
  __syncthreads();

  uint8_t*     Pb = ptrws + (size_t)b * (TMAX * LPAD);
  const float* fb = feats + (size_t)b * (TMAX * LBL);

  // ---------------- forward Viterbi scan (only up to len) ----------------
  float* rd = vit_sh[0];
  float* wr = vit_sh[1];
  for (int t = 0; t < len; ++t) {
    if (t + 16 < len)
      __builtin_prefetch(&fb[(t + 16) * LBL + tosafe], 0, 0);  // global_prefetch_b8

    float best; int bi = argmax_row(rd, tr, &best);

    float logit = fb[t * LBL + tosafe];          // clamped index: always in-bounds
    float vnew  = best + logit;                  // (max + logit)
    if (t == len - 1) vnew = vnew + tstop;       // (+ stop row) exactly like reference

    Pb[(size_t)t * LPAD + to] = (uint8_t)bi;     // backpointer byte (L2-resident ws)

    if (act) wr[to] = vnew;                      // disjoint buffer: no pre-write barrier
    __syncthreads();                             // publish wr before next step reads it

    float* tmp = rd; rd = wr; wr = tmp;
  }
  // final vit now in rd

  // For t >= len vit is frozen => all remaining pointer rows are identical.
  // Compute once, store at row 'len'; backtrack clamps row index to it.
  if (len < TMAX) {
    float best; int bi = argmax_row(rd, tr, &best);
    Pb[(size_t)len * LPAD + to] = (uint8_t)bi;
  }

  // ---------------- final score / argmax ----------------
  if (to == 0) {
    float best = rd[0]; int bi = 0;
    for (int j = 1; j < LBL; ++j) {
      float v = rd[j];
      if (v > best) { best = v; bi = j; }
    }
    out[b] = best;
    idx_sh = bi;
    path_sh[TMAX - 1] = (uint8_t)bi;             // paths[T-1] = idx
  }
  __threadfence();                               // ptr stores visible before async reads
  __syncthreads();

  // ---------------- backtrack: double-buffered async tiles into LDS ----------------
  {
    const int s = NSEG - 1;                      // issue first (highest) segment
    #pragma unroll
    for (int r = 0; r < 8; ++r) {
      int c   = r * 64 + to;                     // 512 chunks of 16B = 8KB tile
      int row = c >> 2;
      int off = (c & 3) * 16;
      int gt  = s * TILE + row;
      int srow = (gt < len) ? gt : len;          // frozen-row clamp
      async_ld_b128((uint32_t)(uintptr_t)&tile_sh[0][row * LPAD + off],
                    srow * LPAD + off,
                    (unsigned long long)(uintptr_t)Pb);
    }
  }

  int idx = idx_sh;
  for (int s = NSEG - 1; s >= 0; --s) {
    const int pcur = (NSEG - 1 - s) & 1;
    wait_async0();                               // s_wait_asynccnt 0
    __syncthreads();

    if (s > 0) {                                 // prefetch next (lower) segment
      const int s2 = s - 1;
      #pragma unroll
      for (int r = 0; r < 8; ++r) {
        int c   = r * 64 + to;
        int row = c >> 2;
        int off = (c & 3) * 16;
        int gt  = s2 * TILE + row;
        int srow = (gt < len) ? gt : len;
        async_ld_b128((uint32_t)(uintptr_t)&tile_sh[pcur ^ 1][row * LPAD + off],
                      srow * LPAD + off,
                      (unsigned long long)(uintptr_t)Pb);
      }
    }

    if (to == 0) {                               // LDS-latency pointer walk
      const uint8_t* tb = tile_sh[pcur];
      const int tlo = (s == 0) ? 1 : s * TILE;   // row 0 is never consumed
      for (int t = s * TILE + TILE - 1; t >= tlo; --t) {
        idx = tb[(t - s * TILE) * LPAD + idx];
        path_sh[t - 1] = (uint8_t)idx;           // paths[t-1] = back_seq[t]
      }
    }
    __syncthreads();
  }

  // ---------------- coalesced path output (harness output dtype: float) ----------------
  float* po = out + BATCH + (size_t)b * TMAX;
  #pragma unroll 4
  for (int t = to; t < TMAX; t += 64) po[t] = (float)path_sh[t];
}

extern "C" void kernel_launch(void* const* d_in, const int* in_sizes, int n_in,
                              void* d_out, int out_size, void* d_ws, size_t ws_size,
                              hipStream_t stream) {
  (void)in_sizes; (void)n_in; (void)out_size; (void)ws_size;
  const float* feats = (const float*)d_in[0];
  const int*   lens  = (const int*)d_in[1];
  const float* trans = (const float*)d_in[2];
  float*   out = (float*)d_out;
  uint8_t* ws  = (uint8_t*)d_ws;   // needs B*T*64 = 32 MB (L2-resident on 192MB L2)

  hipLaunchKernelGGL(crf_viterbi_kernel, dim3(BATCH), dim3(64), 0, stream,
                     feats, lens, trans, out, ws);
}